// ESDFMPCSolver_89300960018673
// MI455X (gfx1250) — compile-verified
//
#include <hip/hip_runtime.h>

typedef float v2f __attribute__((ext_vector_type(2)));
typedef float v8f __attribute__((ext_vector_type(8)));

// async-to-LDS builtin operand types (probe-confirmed: params are pointers to
// int __attribute__((vector_size(8))) in global / LDS address spaces)
typedef int async_i2 __attribute__((vector_size(2 * sizeof(int))));
typedef __attribute__((address_space(1))) async_i2* glb_i2_ptr;
typedef __attribute__((address_space(3))) async_i2* lds_i2_ptr;

#if __has_builtin(__builtin_amdgcn_global_load_async_to_lds_b64) && \
    __has_builtin(__builtin_amdgcn_s_wait_asynccnt)
#define HAVE_ASYNC 1
#define ASYNC_WAIT() __builtin_amdgcn_s_wait_asynccnt(0)
#else
#define HAVE_ASYNC 0
#define ASYNC_WAIT()
#endif

#define CHUNK 2048            // points per LDS buffer (16 KB); double-buffered
#define BLOCK_THREADS 256     // 8 waves (wave32)
#define WAVES_PER_BLOCK 8
#define ROWS_PER_WAVE 16
#define EPS_F 1e-8f
#define SCORE_BIG 3.0e38f

__global__ __launch_bounds__(BLOCK_THREADS) void esdf_wmma_kernel(
        const float2* __restrict__ pc, float* __restrict__ out, int N) {
    __shared__ float2 smem[2][CHUNK];

    const int lane = threadIdx.x & 31;
    // wave id via readfirstlane -> SGPR, so i_base is provably wave-uniform
    const int wave = __builtin_amdgcn_readfirstlane((int)threadIdx.x) >> 5;
    const bool lo = lane < 16;
    const int i_base = (blockIdx.x * WAVES_PER_BLOCK + wave) * ROWS_PER_WAVE;

    // A matrix (16x4 f32): lanes 0-15 hold K0,K1 (M=lane); lanes 16-31 hold K2,K3.
    // A = [-2x_i, -2y_i, 1, 0]  =>  D = |p_j|^2 - 2 p_i . p_j
    float2 pi0 = pc[i_base + (lane & 15)];
    v2f A;
    A[0] = lo ? (-2.0f * pi0.x) : 1.0f;
    A[1] = lo ? (-2.0f * pi0.y) : 0.0f;

    float minv[8];
    int   argj[8];
#pragma unroll
    for (int v = 0; v < 8; ++v) { minv[v] = SCORE_BIG; argj[v] = 0; }

    // ---- stage chunk 0 ----
#if HAVE_ASYNC
#pragma unroll
    for (int k = threadIdx.x; k < CHUNK; k += BLOCK_THREADS)
        __builtin_amdgcn_global_load_async_to_lds_b64(
            (glb_i2_ptr)(pc + k), (lds_i2_ptr)(&smem[0][k]), 0, 0);
#else
    for (int k = threadIdx.x; k < CHUNK; k += BLOCK_THREADS) smem[0][k] = pc[k];
#endif
    ASYNC_WAIT();
    __syncthreads();

    const int nchunks = N / CHUNK;
    for (int c = 0; c < nchunks; ++c) {
        // ---- async-stage next chunk into the other buffer ----
        if (c + 1 < nchunks) {
            const float2* src = pc + (c + 1) * CHUNK;
#if HAVE_ASYNC
#pragma unroll
            for (int k = threadIdx.x; k < CHUNK; k += BLOCK_THREADS)
                __builtin_amdgcn_global_load_async_to_lds_b64(
                    (glb_i2_ptr)(src + k), (lds_i2_ptr)(&smem[(c + 1) & 1][k]), 0, 0);
#else
            for (int k = threadIdx.x; k < CHUNK; k += BLOCK_THREADS)
                smem[(c + 1) & 1][k] = src[k];
#endif
        }

        // ---- compute on current buffer ----
        const float2* buf = smem[c & 1];
        const int jb0 = c * CHUNK;
#pragma unroll 4
        for (int t = 0; t < CHUNK; t += 16) {
            float2 pj = buf[t + (lane & 15)];
            // B (4x16 f32): lanes 0-15: K0=x_j, K1=y_j; lanes 16-31: K2=|p_j|^2, K3=0
            v2f B;
            B[0] = lo ? pj.x : __builtin_fmaf(pj.x, pj.x, pj.y * pj.y);
            B[1] = lo ? pj.y : 0.0f;
            v8f C = {};
            v8f D = __builtin_amdgcn_wmma_f32_16x16x4_f32(
                false, A, false, B, (short)0, C, false, false);

            const int jb = jb0 + t;
            if (jb == i_base) {  // scalar compare (i_base is SGPR); taken once per wave
#pragma unroll
                for (int v = 0; v < 8; ++v) {
                    // D layout: VGPR v -> row v (lanes 0-15) / v+8 (lanes 16-31), col = lane&15
                    bool diag = (lane < 8) ? (v == lane)
                                           : ((lane >= 24) && (v == lane - 24));
                    if (diag) D[v] = SCORE_BIG;
                }
            }
            const int jcol = jb + (lane & 15);
#pragma unroll
            for (int v = 0; v < 8; ++v) {
                if (D[v] < minv[v]) { minv[v] = D[v]; argj[v] = jcol; }
            }
        }

        ASYNC_WAIT();     // this wave's async loads into the other buffer are done
        __syncthreads();  // everyone done computing buf[c&1] and staging buf[(c+1)&1]
    }

    // min+argmin across the 16 lanes of each half-wave (branch-free merges)
#pragma unroll
    for (int off = 8; off >= 1; off >>= 1) {
#pragma unroll
        for (int v = 0; v < 8; ++v) {
            float ov = __shfl_xor(minv[v], off, 32);
            int   oj = __shfl_xor(argj[v], off, 32);
            bool take = (ov < minv[v]) || ((ov == minv[v]) && (oj < argj[v]));
            minv[v] = take ? ov : minv[v];
            argj[v] = take ? oj : argj[v];
        }
    }

    if ((lane & 15) == 0) {  // lane 0 -> rows i_base..+7, lane 16 -> +8..+15
        const int rbase = i_base + ((lane >> 4) << 3);
#pragma unroll
        for (int v = 0; v < 8; ++v) {
            const int i = rbase + v;
            float2 p = pc[i];
            float2 q = pc[argj[v]];
            float sqi  = __builtin_fmaf(p.x, p.x, p.y * p.y);
            float d2   = fmaxf(sqi + minv[v], 0.0f);
            float esdf = sqrtf(d2);
            float dx = p.x - q.x, dy = p.y - q.y;
            float nrm = sqrtf(__builtin_fmaf(dx, dx, dy * dy));
            float inv = 1.0f / (nrm + EPS_F);
            float gx = dx * inv, gy = dy * inv;
            // mu = [gx, -gx, gy, -gy] (4,N); lam = [gx, gy, esdf/10] (3,N)
            out[0 * N + i] = gx;
            out[1 * N + i] = -gx;
            out[2 * N + i] = gy;
            out[3 * N + i] = -gy;
            out[4 * N + i] = gx;
            out[5 * N + i] = gy;
            out[6 * N + i] = esdf * 0.1f;
        }
    }
}

extern "C" void kernel_launch(void* const* d_in, const int* in_sizes, int n_in,
                              void* d_out, int out_size, void* d_ws, size_t ws_size,
                              hipStream_t stream) {
    const float2* pc = (const float2*)d_in[0];
    float* out = (float*)d_out;
    const int N = in_sizes[0] / 2;                             // 16384
    const int blocks = N / (WAVES_PER_BLOCK * ROWS_PER_WAVE);  // 128
    esdf_wmma_kernel<<<blocks, BLOCK_THREADS, 0, stream>>>(pc, out, N);
}